// FastMLNeighborSampler_34359738368605
// MI455X (gfx1250) — compile-verified
//
#include <hip/hip_runtime.h>
#include <hip/hip_bf16.h>

#define NODE_DIM     256
#define MAX_DEG      64
#define NUM_SAMPLES  16
#define GROUP_DIM    4          // MAX_DEG / NUM_SAMPLES
#define ROWS_PER_BLK 16
#define BLOCK_THREADS 256

typedef __attribute__((ext_vector_type(2))) float v2f;
typedef __attribute__((ext_vector_type(8))) float v8f;

// Fused: l = v_f@W + b (fp32 WMMA) -> scores = relu(l . n_f) -> grouped argmin
__global__ __launch_bounds__(BLOCK_THREADS)
void FastMLNeighborSampler_kernel(const int*   __restrict__ ids,
                                  const int*   __restrict__ adj_info,
                                  const float* __restrict__ features,
                                  const float* __restrict__ Wm,
                                  const float* __restrict__ bias,
                                  float* __restrict__ out_sel,
                                  float* __restrict__ out_att,
                                  float* __restrict__ out_nz0,
                                  float* __restrict__ out_nz1,
                                  int n_nodes)
{
    __shared__ int   s_ids[ROWS_PER_BLK];
    __shared__ int   s_adj[ROWS_PER_BLK][MAX_DEG];
    __shared__ float s_l  [ROWS_PER_BLK][NODE_DIM];
    __shared__ float s_sc [ROWS_PER_BLK][MAX_DEG];
    __shared__ float s_nz [ROWS_PER_BLK][NUM_SAMPLES];

    const int tid = threadIdx.x;
    const int v0  = blockIdx.x * ROWS_PER_BLK;

    if (tid < ROWS_PER_BLK) s_ids[tid] = ids[v0 + tid];
    __syncthreads();

    // ---- stage adjacency rows (16 x 64 ints) ----
    for (int i = tid; i < ROWS_PER_BLK * MAX_DEG; i += BLOCK_THREADS) {
        const int r = i >> 6;           // / MAX_DEG
        const int c = i & (MAX_DEG - 1);
        s_adj[r][c] = adj_info[(size_t)s_ids[r] * MAX_DEG + c];
    }

    // ---- GEMM phase: l[16 x 256] = v_f @ W, fp32 WMMA 16x16x4 ----
    {
        const int wave = tid >> 5;
        const int lane = tid & 31;
        const int half = lane >> 4;     // 0: lanes 0-15, 1: lanes 16-31
        const int mrow = lane & 15;

        // A fragment: lane holds v_f[mrow][k0 + 2*half], v_f[mrow][k0 + 2*half + 1]
        const float* arow = features + (size_t)s_ids[mrow] * NODE_DIM;

        const int col0 = (wave * 2) * 16 + (lane & 15);   // first column tile
        const int col1 = col0 + 16;                       // second column tile

        v8f acc0 = {};
        v8f acc1 = {};

        #pragma unroll 4
        for (int k0 = 0; k0 < NODE_DIM; k0 += 4) {
            const int ka = k0 + 2 * half;

            v2f a = *(const v2f*)(arow + ka);   // 8B aligned (ka even)

            v2f b0, b1;
            b0.x = Wm[(size_t)ka * NODE_DIM + col0];
            b0.y = Wm[(size_t)(ka + 1) * NODE_DIM + col0];
            b1.x = Wm[(size_t)ka * NODE_DIM + col1];
            b1.y = Wm[(size_t)(ka + 1) * NODE_DIM + col1];

            acc0 = __builtin_amdgcn_wmma_f32_16x16x4_f32(
                       false, a, false, b0, (short)0, acc0, false, false);
            acc1 = __builtin_amdgcn_wmma_f32_16x16x4_f32(
                       false, a, false, b1, (short)0, acc1, false, false);
        }

        const float bv0 = bias[col0];
        const float bv1 = bias[col1];
        // C layout: VGPR j -> row (half*8 + j), col = col0 + (lane&15)
        #pragma unroll
        for (int j = 0; j < 8; ++j) {
            const int m = half * 8 + j;
            s_l[m][col0] = acc0[j] + bv0;
            s_l[m][col1] = acc1[j] + bv1;
        }
    }
    __syncthreads();

    // ---- scores phase: 16*64 dots of length 256 (4 per thread) ----
    for (int p = tid; p < ROWS_PER_BLK * MAX_DEG; p += BLOCK_THREADS) {
        const int v = p >> 6;
        const int k = p & (MAX_DEG - 1);
        const int nbr = s_adj[v][k];
        const float* nf = features + (size_t)nbr * NODE_DIM;

        float dot = 0.0f;
        #pragma unroll 8
        for (int d = 0; d < NODE_DIM; d += 4) {
            const float4 f = *(const float4*)(nf + d);
            dot = fmaf(s_l[v][d + 0], f.x,
                  fmaf(s_l[v][d + 1], f.y,
                  fmaf(s_l[v][d + 2], f.z,
                  fmaf(s_l[v][d + 3], f.w, dot))));
        }
        s_sc[v][k] = fmaxf(dot, 0.0f);   // relu
    }
    __syncthreads();

    // ---- grouped argmin + outputs: one (v, sample) per thread ----
    {
        const int v = tid >> 4;             // 0..15
        const int s = tid & (NUM_SAMPLES - 1);
        const float* g = &s_sc[v][s * GROUP_DIM];

        int   best = 0;
        float bmin = g[0];
        #pragma unroll
        for (int j = 1; j < GROUP_DIM; ++j) {
            if (g[j] < bmin) { bmin = g[j]; best = j; }   // first-min tie-break
        }
        const int selnode = s_adj[v][s * GROUP_DIM + best];
        const int vg = v0 + v;

        out_sel[(size_t)vg * NUM_SAMPLES + s] = (float)selnode;
        out_att[(size_t)vg * NUM_SAMPLES + s] = 1.0f;
        s_nz[v][s] = (selnode == n_nodes - 1) ? 0.0f : 1.0f;
    }
    __syncthreads();

    if (tid < ROWS_PER_BLK) {
        float sum = 0.0f;
        #pragma unroll
        for (int s = 0; s < NUM_SAMPLES; ++s) sum += s_nz[tid][s];
        out_nz0[v0 + tid] = sum;
        out_nz1[v0 + tid] = sum;
    }
}

extern "C" void kernel_launch(void* const* d_in, const int* in_sizes, int n_in,
                              void* d_out, int out_size, void* d_ws, size_t ws_size,
                              hipStream_t stream) {
    const int*   ids      = (const int*)  d_in[0];
    const int*   adj_info = (const int*)  d_in[1];
    const float* features = (const float*)d_in[2];
    const float* Wm       = (const float*)d_in[3];
    const float* bias     = (const float*)d_in[4];
    // d_in[5] = num_samples (compile-time 16 here)

    const int V       = in_sizes[0];
    const int n_nodes = in_sizes[1] / MAX_DEG;

    float* o = (float*)d_out;
    float* out_sel = o;                                   // [V,16]
    float* out_att = o + (size_t)V * NUM_SAMPLES;         // [V,16]
    float* out_nz0 = o + (size_t)2 * V * NUM_SAMPLES;     // [V]
    float* out_nz1 = out_nz0 + V;                         // [V]

    const int grid = V / ROWS_PER_BLK;                    // V=8192 -> 512 blocks
    FastMLNeighborSampler_kernel<<<grid, BLOCK_THREADS, 0, stream>>>(
        ids, adj_info, features, Wm, bias,
        out_sel, out_att, out_nz0, out_nz1, n_nodes);
}